// MySoftmax_74672301408351
// MI455X (gfx1250) — compile-verified
//
#include <hip/hip_runtime.h>
#include <hip/hip_bf16.h>
#include <stdint.h>

// Softmax over H*W=65536 per (b,c) row. One workgroup per row; the whole
// 256KB row is staged in LDS (CDNA5: 320KB/WGP) via async global->LDS copies
// so HBM sees exactly one read + one write (roofline floor ~22us @ 23.3TB/s).
// The max pass is pipelined against the async DMA using descending
// s_wait_asynccnt values (async loads retire in order per wave, and each
// thread consumes exactly the chunks its own wave's k-th async op wrote).

#define ROW_LEN 65536            // H*W elements per row
#define NCHUNK  (ROW_LEN / 4)    // 16384 float4 chunks
#define BLOCK   1024             // 32 waves (wave32)
#define PER_T   (NCHUNK / BLOCK) // 16 chunks per thread

// Native clang vector: valid for __builtin_nontemporal_store (HIP float4 is not).
typedef __attribute__((ext_vector_type(4))) float f32x4;

// s_wait_asynccnt needs a literal SIMM16; this switch constant-folds when the
// surrounding loop is fully unrolled.
__device__ __forceinline__ void wait_asynccnt_le(int n) {
  switch (n) {
    case 0:  asm volatile("s_wait_asynccnt 0"  ::: "memory"); break;
    case 1:  asm volatile("s_wait_asynccnt 1"  ::: "memory"); break;
    case 2:  asm volatile("s_wait_asynccnt 2"  ::: "memory"); break;
    case 3:  asm volatile("s_wait_asynccnt 3"  ::: "memory"); break;
    case 4:  asm volatile("s_wait_asynccnt 4"  ::: "memory"); break;
    case 5:  asm volatile("s_wait_asynccnt 5"  ::: "memory"); break;
    case 6:  asm volatile("s_wait_asynccnt 6"  ::: "memory"); break;
    case 7:  asm volatile("s_wait_asynccnt 7"  ::: "memory"); break;
    case 8:  asm volatile("s_wait_asynccnt 8"  ::: "memory"); break;
    case 9:  asm volatile("s_wait_asynccnt 9"  ::: "memory"); break;
    case 10: asm volatile("s_wait_asynccnt 10" ::: "memory"); break;
    case 11: asm volatile("s_wait_asynccnt 11" ::: "memory"); break;
    case 12: asm volatile("s_wait_asynccnt 12" ::: "memory"); break;
    case 13: asm volatile("s_wait_asynccnt 13" ::: "memory"); break;
    case 14: asm volatile("s_wait_asynccnt 14" ::: "memory"); break;
    case 15: asm volatile("s_wait_asynccnt 15" ::: "memory"); break;
    default: break;
  }
}

__device__ __forceinline__ float wave_red(float v, bool is_max) {
#pragma unroll
  for (int off = 16; off > 0; off >>= 1) {
    float o = __shfl_xor(v, off, 32);
    v = is_max ? fmaxf(v, o) : (v + o);
  }
  return v;
}

// Block-wide reduction over 32 waves; `red` is a 32-float LDS scratch array.
__device__ __forceinline__ float block_red(float v, float* red, int tid, bool is_max) {
  v = wave_red(v, is_max);
  const int wave = tid >> 5;
  const int lane = tid & 31;
  __syncthreads();                 // protect red[] reuse across calls
  if (lane == 0) red[wave] = v;
  __syncthreads();
  if (wave == 0) {
    float x = red[lane];           // exactly 32 partials
    x = wave_red(x, is_max);
    if (lane == 0) red[0] = x;
  }
  __syncthreads();
  return red[0];
}

extern "C" __global__ void __launch_bounds__(BLOCK, 1)
softmax_row_kernel(const float* __restrict__ x, float* __restrict__ out) {
  extern __shared__ __align__(16) float smem[];   // [ROW_LEN] data + [32] partials
  f32x4* smem4 = (f32x4*)smem;
  float* red   = smem + ROW_LEN;

  const int tid = threadIdx.x;
  const long long row = blockIdx.x;
  const uint64_t gsrc = (uint64_t)(x   + row * (long long)ROW_LEN);
  float*         gdst =            out + row * (long long)ROW_LEN;

  // ---- Phase 0: async DMA the whole row into LDS (no VGPR staging) ----
  // Each lane moves a 16B chunk -> 512B contiguous per wave instruction.
#pragma unroll
  for (int k = 0; k < PER_T; ++k) {
    const unsigned c       = (unsigned)(tid + k * BLOCK);
    const unsigned lds_off = c * 16u;           // byte offset into LDS alloc
    const uint64_t gaddr   = gsrc + (uint64_t)c * 16u;
    asm volatile("global_load_async_to_lds_b128 %0, %1, off"
                 :: "v"(lds_off), "v"(gaddr)
                 : "memory");
  }

  // ---- Phase 1: row max, pipelined against the in-flight DMA ----
  // Thread tid only touches chunks its own wave's k-th async op wrote, and
  // async loads complete in order, so iteration k needs asynccnt <= 15-k.
  float m = -__builtin_inff();
#pragma unroll
  for (int k = 0; k < PER_T; ++k) {
    wait_asynccnt_le(PER_T - 1 - k);
    f32x4 v = smem4[tid + k * BLOCK];
    m = fmaxf(m, fmaxf(fmaxf(v.x, v.y), fmaxf(v.z, v.w)));
  }
  const float M = block_red(m, red, tid, /*is_max=*/true);

  // ---- Phase 2: e = exp(x - M), written back in place; accumulate sum ----
  float s = 0.0f;
#pragma unroll
  for (int k = 0; k < PER_T; ++k) {
    const int c = tid + k * BLOCK;
    f32x4 v = smem4[c];
    v.x = __expf(v.x - M);
    v.y = __expf(v.y - M);
    v.z = __expf(v.z - M);
    v.w = __expf(v.w - M);
    smem4[c] = v;
    s += (v.x + v.y) + (v.z + v.w);
  }
  const float S = block_red(s, red, tid, /*is_max=*/false);
  const float r = 1.0f / S;

  // ---- Phase 3: normalize; stream out non-temporal (never re-read) ----
#pragma unroll
  for (int k = 0; k < PER_T; ++k) {
    const int c = tid + k * BLOCK;
    f32x4 v = smem4[c];
    v.x *= r; v.y *= r; v.z *= r; v.w *= r;
    __builtin_nontemporal_store(v, (f32x4*)(gdst + (size_t)c * 4));
  }
}

extern "C" void kernel_launch(void* const* d_in, const int* in_sizes, int n_in,
                              void* d_out, int out_size, void* d_ws, size_t ws_size,
                              hipStream_t stream) {
  (void)n_in; (void)d_ws; (void)ws_size; (void)out_size;
  const float* x = (const float*)d_in[0];
  float* out = (float*)d_out;

  const int rows = in_sizes[0] / ROW_LEN;                 // B*C = 1024
  const size_t shmem = (ROW_LEN + 32) * sizeof(float);    // 256KB + 128B <= 320KB/WGP

  // Raise the dynamic-LDS cap (deterministic, not a stream op: capture-safe).
  (void)hipFuncSetAttribute(reinterpret_cast<const void*>(softmax_row_kernel),
                            hipFuncAttributeMaxDynamicSharedMemorySize, (int)shmem);

  softmax_row_kernel<<<rows, BLOCK, shmem, stream>>>(x, out);
}